// CSILoss_74156905332744
// MI455X (gfx1250) — compile-verified
//
#include <hip/hip_runtime.h>
#include <math.h>
#include <stdint.h>

#define TPB    256          // 8 wave32s per block
#define NCOLS  4096
#define CHUNK  1024         // elements per LDS tile per array
#define NCHUNK (NCOLS / CHUNK)

#define EPS_C  1e-12f
#define EPS_D  1e-8f
#define MAXM   100.0f

typedef __attribute__((ext_vector_type(2))) float v2f;
typedef __attribute__((ext_vector_type(8))) float v8f;

// Single shared object per kernel -> LDS offset 0; async dest addresses are
// computed as byte offsets into this struct.
struct __align__(16) SMem {
  float sh[2][4][CHUNK];   // [buffer][array: pr,pi,tr,ti][elem]  = 32 KB
  float red[8][16];        // cross-wave reduction scratch (rows 9..15 zeroed)
};

// ---- CDNA5 async global->LDS copy (ASYNCcnt-tracked), GV addressing ----
__device__ __forceinline__ void async_ld_b128(unsigned lds_byte_off, const float* gaddr) {
  asm volatile("global_load_async_to_lds_b128 %0, %1, off"
               :: "v"(lds_byte_off), "v"(gaddr)
               : "memory");
}
__device__ __forceinline__ void wait_async0() {
  asm volatile("s_wait_asynccnt 0x0" ::: "memory");
}

// wave32 butterfly sum
__device__ __forceinline__ float wred(float v) {
#pragma unroll
  for (int o = 16; o > 0; o >>= 1) v += __shfl_xor(v, o, 32);
  return v;
}

// ---- f32-exact cross-wave reduction on the matrix pipe ---------------------
// red[8][16] holds 8 wave-partials for up to 16 quantities. Executed by a
// fully-active wave32: two chained V_WMMA_F32_16X16X4_F32 (C-accumulate)
// produce D[m,n] = sum over 8 waves of red[w][m]. Result placement per the
// 16x16 f32 C/D layout: lane 0 VGPR r = total[r] (r=0..7), lane 16 VGPR r =
// total[8+r].
__device__ __forceinline__ v8f wmma_wave_totals(const float (*red)[16], int lane) {
  const int m  = lane & 15;
  const int hi = (lane >> 4) & 1;            // 0: K0/K1 (waves 0,1 | 2,3) halves
  v2f a1, a2, b;
  a1.x = red[hi ? 2 : 0][m];                 // A[m, K0] / A[m, K2]
  a1.y = red[hi ? 3 : 1][m];                 // A[m, K1] / A[m, K3]
  a2.x = red[hi ? 6 : 4][m];
  a2.y = red[hi ? 7 : 5][m];
  b.x = 1.0f; b.y = 1.0f;                    // B = ones(4x16)
  v8f c = {};
  c = __builtin_amdgcn_wmma_f32_16x16x4_f32(false, a1, false, b, (short)0, c, false, false);
  c = __builtin_amdgcn_wmma_f32_16x16x4_f32(false, a2, false, b, (short)0, c, false, false);
  return c;
}

__device__ __forceinline__ void accum_elem(
    float pr, float pim, float tr, float ti,
    float& s_pm, float& s_pm2, float& s_tm, float& s_tm2,
    float& s_a, float& s_b, float& s_mag, float& s_ph, float& s_cor) {
  float pm_raw = sqrtf(pr * pr + pim * pim);
  float tm_raw = sqrtf(tr * tr + ti * ti);
  float pm = fminf(fmaxf(pm_raw, EPS_C), MAXM);
  float tm = fminf(fmaxf(tm_raw, EPS_C), MAXM);
  float d = pm - tm;
  s_mag += d * d;
  s_pm += pm;  s_pm2 += pm * pm;
  s_tm += tm;  s_tm2 += tm * tm;
  s_a += pm_raw; s_b += tm_raw;
  // wrapped phase difference == angle(p * conj(t)) with the +eps shifts
  float pre = pr + EPS_D, tre = tr + EPS_D;
  float num = pim * tre - pre * ti;
  float den = pre * tre + pim * ti;
  float pd = atan2f(num, den);
  s_ph += pd * pd;
  float pinv = 1.0f / (pm + EPS_D);
  float tinv = 1.0f / (tm + EPS_D);
  float dre = pr * pinv - tr * tinv;
  float dim = pim * pinv - ti * tinv;
  s_cor += dre * dre + dim * dim;
}

// -------- Pass 1: per-row sums + global-loss partials (async-LDS staged) ----
__global__ __launch_bounds__(TPB) void pass1_kernel(
    const float* __restrict__ pr, const float* __restrict__ pim,
    const float* __restrict__ tr, const float* __restrict__ ti,
    float* __restrict__ ws, int rows) {
  __shared__ SMem smem;
  const int row = blockIdx.x;
  const int t   = threadIdx.x;
  const size_t rb = (size_t)row * NCOLS;
  const float* g[4] = { pr + rb, pim + rb, tr + rb, ti + rb };

  float s_pm = 0.f, s_pm2 = 0.f, s_tm = 0.f, s_tm2 = 0.f;
  float s_a = 0.f, s_b = 0.f, s_mag = 0.f, s_ph = 0.f, s_cor = 0.f;

  const unsigned lane_off = (unsigned)t * 16u;   // B128 per lane per array

  // prime buffer 0 with chunk 0
#pragma unroll
  for (int k = 0; k < 4; ++k)
    async_ld_b128((unsigned)((0 * 4 + k) * CHUNK * 4) + lane_off, g[k] + t * 4);

  for (int c = 0; c < NCHUNK; ++c) {
    wait_async0();       // this wave's 4 outstanding async loads (chunk c) done
    __syncthreads();     // all waves have chunk c; all finished compute(c-1)
    if (c + 1 < NCHUNK) {
      const int nb = (c + 1) & 1;
#pragma unroll
      for (int k = 0; k < 4; ++k)
        async_ld_b128((unsigned)((nb * 4 + k) * CHUNK * 4) + lane_off,
                      g[k] + (size_t)(c + 1) * CHUNK + t * 4);
    }
    const int buf = c & 1;
    float4 vpr = *(const float4*)&smem.sh[buf][0][t * 4];
    float4 vpi = *(const float4*)&smem.sh[buf][1][t * 4];
    float4 vtr = *(const float4*)&smem.sh[buf][2][t * 4];
    float4 vti = *(const float4*)&smem.sh[buf][3][t * 4];
    accum_elem(vpr.x, vpi.x, vtr.x, vti.x, s_pm, s_pm2, s_tm, s_tm2, s_a, s_b, s_mag, s_ph, s_cor);
    accum_elem(vpr.y, vpi.y, vtr.y, vti.y, s_pm, s_pm2, s_tm, s_tm2, s_a, s_b, s_mag, s_ph, s_cor);
    accum_elem(vpr.z, vpi.z, vtr.z, vti.z, s_pm, s_pm2, s_tm, s_tm2, s_a, s_b, s_mag, s_ph, s_cor);
    accum_elem(vpr.w, vpi.w, vtr.w, vti.w, s_pm, s_pm2, s_tm, s_tm2, s_a, s_b, s_mag, s_ph, s_cor);
  }

  float vals[9] = { s_pm, s_pm2, s_tm, s_tm2, s_a, s_b, s_mag, s_ph, s_cor };
  const int lane = t & 31, wv = t >> 5;
#pragma unroll
  for (int i = 0; i < 9; ++i) vals[i] = wred(vals[i]);
  if (lane == 0) {
#pragma unroll
    for (int i = 0; i < 16; ++i) smem.red[wv][i] = (i < 9) ? vals[i] : 0.0f;
  }
  __syncthreads();
  if (t < 32) {                      // wave 0, EXEC all-ones for WMMA
    v8f c = wmma_wave_totals(smem.red, lane);
    if (lane == 0) {
#pragma unroll
      for (int i = 0; i < 8; ++i) ws[(size_t)i * rows + row] = c[i];
    }
    if (lane == 16) ws[(size_t)8 * rows + row] = c[0];   // quantity 8
  }
}

// -------- Pass 2: per-row Jensen-Shannon (reversed rows -> L2 hits) --------
__global__ __launch_bounds__(TPB) void pass2_kernel(
    const float* __restrict__ pr, const float* __restrict__ pim,
    const float* __restrict__ tr, const float* __restrict__ ti,
    float* __restrict__ ws, int rows) {
  __shared__ float red2[8];
  const int row = (int)gridDim.x - 1 - (int)blockIdx.x;  // tail rows are L2-resident
  const int t = threadIdx.x;
  const float A  = ws[(size_t)4 * rows + row] + (float)NCOLS * EPS_C;
  const float Bs = ws[(size_t)5 * rows + row] + (float)NCOLS * EPS_C;
  const float invA = 1.0f / A, invB = 1.0f / Bs;
  const size_t rb = (size_t)row * NCOLS;

  float acc = 0.f;
#pragma unroll
  for (int c = 0; c < NCOLS / (TPB * 4); ++c) {
    const size_t o = rb + (size_t)c * TPB * 4 + (size_t)t * 4;
    float4 vpr = *(const float4*)(pr + o);
    float4 vpi = *(const float4*)(pim + o);
    float4 vtr = *(const float4*)(tr + o);
    float4 vti = *(const float4*)(ti + o);
#pragma unroll
    for (int e = 0; e < 4; ++e) {
      float xr = (&vpr.x)[e], xi = (&vpi.x)[e];
      float yr = (&vtr.x)[e], yi = (&vti.x)[e];
      float pm = sqrtf(xr * xr + xi * xi);
      float tm = sqrtf(yr * yr + yi * yi);
      float p = (pm + EPS_C) * invA;
      float q = (tm + EPS_C) * invB;
      float im2 = 2.0f / (p + q);                 // 1/m scaled
      acc += p * __logf(p * im2) + q * __logf(q * im2);
    }
  }
  acc = wred(acc);
  if ((t & 31) == 0) red2[t >> 5] = acc;
  __syncthreads();
  if (t == 0) {
    float s = 0.f;
    for (int w = 0; w < TPB / 32; ++w) s += red2[w];
    ws[(size_t)9 * rows + row] = 0.5f * s;
  }
}

// -------- Pass 3: combine 4096 row records into the scalar loss ------------
__global__ __launch_bounds__(TPB) void finalize_kernel(
    const float* __restrict__ ws, float* __restrict__ out, int rows) {
  __shared__ float red[8][16];
  const int t = threadIdx.x;
  float g_mag = 0.f, g_ph = 0.f, g_cor = 0.f, g_mean = 0.f, g_std = 0.f, g_js = 0.f;
  const float invN = 1.0f / (float)NCOLS;
  for (int r = t; r < rows; r += TPB) {
    float spm  = ws[(size_t)0 * rows + r];
    float spm2 = ws[(size_t)1 * rows + r];
    float stm  = ws[(size_t)2 * rows + r];
    float stm2 = ws[(size_t)3 * rows + r];
    g_mag += ws[(size_t)6 * rows + r];
    g_ph  += ws[(size_t)7 * rows + r];
    g_cor += ws[(size_t)8 * rows + r];
    g_js  += ws[(size_t)9 * rows + r];
    float pmean = spm * invN, tmean = stm * invN;
    float dm = pmean - tmean;
    g_mean += dm * dm;
    float pvar = fmaxf(spm2 * invN - pmean * pmean, EPS_C);
    float tvar = fmaxf(stm2 * invN - tmean * tmean, EPS_C);
    float dsd = sqrtf(pvar) - sqrtf(tvar);
    g_std += dsd * dsd;
  }
  float vals[6] = { g_mag, g_ph, g_cor, g_mean, g_std, g_js };
  const int lane = t & 31, wv = t >> 5;
#pragma unroll
  for (int i = 0; i < 6; ++i) vals[i] = wred(vals[i]);
  if (lane == 0) {
#pragma unroll
    for (int i = 0; i < 16; ++i) red[wv][i] = (i < 6) ? vals[i] : 0.0f;
  }
  __syncthreads();
  if (t < 32) {                      // wave 0, EXEC all-ones for WMMA
    v8f c = wmma_wave_totals(red, lane);
    if (lane == 0) {
      // quantities 0..5 all land in lane 0's D VGPRs
      const float invBN = 1.0f / ((float)rows * (float)NCOLS);
      const float invBr = 1.0f / (float)rows;
      float mag_loss   = c[0] * invBN;
      float phase_loss = c[1] * invBN;
      float corr_loss  = c[2] * invBN;
      float mean_loss  = c[3] * invBr;
      float std_loss   = c[4] * invBr;
      float js_loss    = c[5] * invBr;
      out[0] = 0.5f * mag_loss + 0.25f * mean_loss + 0.15f * std_loss
             + 0.5f * phase_loss + 0.2f * corr_loss + 0.1f * js_loss;
    }
  }
}

extern "C" void kernel_launch(void* const* d_in, const int* in_sizes, int n_in,
                              void* d_out, int out_size, void* d_ws, size_t ws_size,
                              hipStream_t stream) {
  (void)n_in; (void)out_size; (void)ws_size;
  const float* pr  = (const float*)d_in[0];
  const float* pim = (const float*)d_in[1];
  const float* tr  = (const float*)d_in[2];
  const float* ti  = (const float*)d_in[3];
  float* ws  = (float*)d_ws;   // 10 * rows floats, fully rewritten each call
  float* out = (float*)d_out;
  const int rows = in_sizes[0] / NCOLS;

  pass1_kernel<<<rows, TPB, 0, stream>>>(pr, pim, tr, ti, ws, rows);
  pass2_kernel<<<rows, TPB, 0, stream>>>(pr, pim, tr, ti, ws, rows);
  finalize_kernel<<<1, TPB, 0, stream>>>(ws, out, rows);
}